// UDCTModule_19404662243894
// MI455X (gfx1250) — compile-verified
//
#include <hip/hip_runtime.h>

// ---------------------------------------------------------------------------
// UDCT forward for 1024x1024, cfg=[[3,3],[6,6]], overlap=0.15  (gfx1250)
//
//   1) 2-D forward FFT (rows, then cols) in f32 complex. Radix-16 stages are
//      V_WMMA_F32_16X16X4_F32 matmuls against the DFT-16 matrix; one wave
//      owns one 1-D transform staged in its private LDS slice (explicit
//      address_space(3) pointers -> native ds_load/ds_store, 32-bit addrs).
//   2) Bulk data movement uses the Tensor Data Mover:
//      tensor_load_to_lds / tensor_store_from_lds with 1xN (strided column)
//      or Nx1 (contiguous row) tile descriptors, synced via s_wait_tensorcnt.
//   3) Per band: analytic curvelet window * spectrum, alias-folded onto the
//      decimated grid, then a small inverse 2-D FFT (N=256/512/1024); real
//      part scaled by 1/(N*N*d) written to the band's output slot.
// ---------------------------------------------------------------------------

typedef float v2f __attribute__((ext_vector_type(2)));
typedef float v8f __attribute__((ext_vector_type(8)));
typedef unsigned int u32x4 __attribute__((ext_vector_type(4)));
typedef unsigned int u32x8 __attribute__((ext_vector_type(8)));

// LDS-resident complex: explicit AS(3) so accesses lower to ds_* ops even
// though the pointers also feed ptrtoint for the TDM descriptors.
typedef __attribute__((address_space(3))) v2f lds_v2f;

static __device__ __forceinline__ lds_v2f* to_lds(v2f* p) {
  return (lds_v2f*)(unsigned)(uintptr_t)p;     // low 32 bits == LDS offset
}

#define PI_F     3.14159265358979323846f
#define TWOPI_F  6.28318530717958647692f
#define OVL_F    0.15f
#define A0_F     (PI_F * 0.25f)   // pi/4
#define A1_F     (PI_F * 0.5f)    // pi/2

static __device__ __forceinline__ v8f wmma4(v2f a, v2f b, v8f c) {
  // D = A(16x4) * B(4x16) + C(16x16), all f32, wave32.
  return __builtin_amdgcn_wmma_f32_16x16x4_f32(false, a, false, b,
                                               (short)0, c, false, false);
}

// ------------------------- Tensor Data Mover helpers -----------------------
// D# for a tile_w x tile_h tile of 8-byte elements with row pitch `pitch`
// (all in elements). Group0: count=1, lds_addr, 57-bit global addr, type=2.
// Group1: data_size=3 (8B), tensor dims = tile dims (no OOB clipping),
// tensor_dim0_stride = pitch.  Groups 2/3 omitted (2-D form).

static __device__ __forceinline__ void tdm_make_desc(u32x4& g0, u32x8& g1,
                                                     const void* gbase,
                                                     unsigned lds_off,
                                                     int tile_w, int tile_h,
                                                     int pitch) {
  unsigned long long ga = (unsigned long long)(uintptr_t)gbase;
  g0[0] = 1u;                                                 // count=1
  g0[1] = lds_off;                                            // lds_addr bytes
  g0[2] = (unsigned)ga;                                       // global[31:0]
  g0[3] = (unsigned)((ga >> 32) & 0x01FFFFFFu) | (2u << 30);  // [56:32]|type=2
  g1[0] = 3u << 16;                                           // data_size=8B
  g1[1] = ((unsigned)tile_w & 0xFFFFu) << 16;                 // tensor_dim0 lo
  g1[2] = (((unsigned)tile_w >> 16) & 0xFFFFu)                // tensor_dim0 hi
        | (((unsigned)tile_h & 0xFFFFu) << 16);               // tensor_dim1 lo
  g1[3] = (((unsigned)tile_h >> 16) & 0xFFFFu)                // tensor_dim1 hi
        | (((unsigned)tile_w & 0xFFFFu) << 16);               // tile_dim0
  g1[4] = (unsigned)tile_h & 0xFFFFu;                         // tile_dim1
  g1[5] = (unsigned)pitch;                                    // dim0 stride
  g1[6] = 0u;
  g1[7] = 0u;
}

static __device__ __forceinline__ void tdm_load(const void* gbase,
                                                lds_v2f* lds_ptr,
                                                int tile_w, int tile_h,
                                                int pitch) {
  u32x4 g0; u32x8 g1;
  tdm_make_desc(g0, g1, gbase, (unsigned)(uintptr_t)lds_ptr,
                tile_w, tile_h, pitch);
  asm volatile("tensor_load_to_lds %0, %1" :: "s"(g0), "s"(g1) : "memory");
}

static __device__ __forceinline__ void tdm_store(void* gbase,
                                                 lds_v2f* lds_ptr,
                                                 int tile_w, int tile_h,
                                                 int pitch) {
  u32x4 g0; u32x8 g1;
  tdm_make_desc(g0, g1, gbase, (unsigned)(uintptr_t)lds_ptr,
                tile_w, tile_h, pitch);
  asm volatile("tensor_store_from_lds %0, %1" :: "s"(g0), "s"(g1) : "memory");
}

static __device__ __forceinline__ void wait_tensor0() {
  __builtin_amdgcn_s_wait_tensorcnt(0);
}

static __device__ __forceinline__ void wait_ds0() {
  asm volatile("s_wait_dscnt 0x0" ::: "memory");
}

// ------------------------------ window math --------------------------------

static __device__ __forceinline__ float rampf(float t) {
  t = fminf(fmaxf(t, 0.0f), 1.0f);
  float t2 = t * t;
  return t2 * t2 * (35.0f + t * (-84.0f + t * (70.0f - 20.0f * t)));
}

static __device__ __forceinline__ float risef(float r, float a) {
  return rampf((r - a * (1.0f - OVL_F)) / (2.0f * OVL_F * a));
}

// scale: 0 = lowpass, 1 = first wedge scale (M=6), 2 = second (M=12)
static __device__ float winval(int scale, int m, int iy, int ix) {
  float ky = TWOPI_F * (float)((iy < 512) ? iy : iy - 1024) * (1.0f / 1024.0f);
  float kx = TWOPI_F * (float)((ix < 512) ? ix : ix - 1024) * (1.0f / 1024.0f);
  float r = sqrtf(kx * kx + ky * ky);
  if (scale == 0) {
    return sqrtf(fmaxf(1.0f - risef(r, A0_F), 0.0f));
  }
  float rad;
  int M;
  if (scale == 1) {
    rad = risef(r, A0_F) * (1.0f - risef(r, A1_F));
    M = 6;
  } else {
    rad = risef(r, A1_F);
    M = 12;
  }
  if (rad <= 0.0f) return 0.0f;   // compact radial support: skip atan2 etc.
  float th = atan2f(ky, kx);
  th = th - floorf(th * (1.0f / PI_F)) * PI_F;          // mod pi
  float Mf = (float)M;
  float hw = PI_F / (2.0f * Mf);
  float ow = OVL_F * hw;
  float c  = PI_F * ((float)m + 0.5f) / Mf;
  float d  = th - c + 0.5f * PI_F;
  d = d - floorf(d * (1.0f / PI_F)) * PI_F - 0.5f * PI_F;
  float g = 1.0f - rampf((fabsf(d) - (hw - ow)) / (2.0f * ow));
  return sqrtf(fmaxf(rad * g, 0.0f));
}

// --------------------- wave-level FFT, N = 16*16*R -------------------------
// One wave owns one transform. Buffers are interleaved complex in LDS.
// Input x[n] in b0[n], output X[k] in b1[k]. SIGN=-1 fwd, +1 inverse.
//
// n = 16R*n1 + R*n2 + n3 ; k = k1 + 16*k2 + 256*k3
// Stage A: Y[k1][n2*R+n3] = sum_n1 W16[k1][n1] x[..] * w_{16R}^{n2 k1}
// Stage B: Z[k2][k1*R+n3] = sum_n2 W16[k2][n2] Y[..] * w_{N}^{n3(k1+16k2)}
// Stage C: radix-R butterflies over n3.

template <int R, int SIGN>
static __device__ __forceinline__ void wave_fft(lds_v2f* b0, lds_v2f* b1) {
  static_assert(R == 1 || R == 2 || R == 4, "bad radix");
  constexpr int NC = 16 * R;     // columns of the 16 x NC stage matrices
  constexpr int N  = 256 * R;
  const int lane = (int)(threadIdx.x & 31u);
  const int half = lane >> 4;
  const int lcol = lane & 15;
  const float sgn = (float)SIGN;

  // DFT-16 matrix in WMMA A-operand layout (16x4 chunk j):
  //   lane L: M = L%16 ; VGPR0 holds K = 4j+2*(L/16), VGPR1 holds K+1.
  v2f ar[4], ai[4];
#pragma unroll
  for (int j = 0; j < 4; ++j) {
    int K0 = 4 * j + 2 * half;
    float s0, c0, s1, c1;
    __sincosf(sgn * (TWOPI_F / 16.0f) * (float)((lcol * K0) & 15), &s0, &c0);
    __sincosf(sgn * (TWOPI_F / 16.0f) * (float)((lcol * (K0 + 1)) & 15), &s1, &c1);
    v2f tr; tr.x = c0; tr.y = c1; ar[j] = tr;
    v2f ti; ti.x = s0; ti.y = s1; ai[j] = ti;
  }

  // ---------------- stage A : b0 -> b1 ----------------
#pragma unroll
  for (int t = 0; t < R; ++t) {
    const int c = 16 * t + lcol;            // column n2*R + n3
    v2f br[4], bi[4];
#pragma unroll
    for (int j = 0; j < 4; ++j) {
      int K0 = 4 * j + 2 * half;            // row n1 (B layout mirrors A)
      const v2f f0 = b0[K0 * NC + c];
      const v2f f1 = b0[(K0 + 1) * NC + c];
      v2f tr; tr.x = f0.x; tr.y = f1.x; br[j] = tr;
      v2f ti; ti.x = f0.y; ti.y = f1.y; bi[j] = ti;
    }
    v8f crr = {}; v8f cii = {}; v8f cim = {};
#pragma unroll
    for (int j = 0; j < 4; ++j) {
      crr = wmma4(ar[j], br[j], crr);       // Re*Re
      cii = wmma4(ai[j], bi[j], cii);       // Im*Im
      cim = wmma4(ar[j], bi[j], cim);       // Re*Im
      cim = wmma4(ai[j], br[j], cim);       // + Im*Re
    }
#pragma unroll
    for (int g = 0; g < 8; ++g) {
      const int M  = g + 8 * half;          // k1 (C/D layout)
      const int n2 = c / R;
      const int e  = (n2 * M) % NC;
      float s, co; __sincosf(sgn * (TWOPI_F / (float)NC) * (float)e, &s, &co);
      const float yr = crr[g] - cii[g];
      const float yi = cim[g];
      v2f o; o.x = yr * co - yi * s; o.y = yr * s + yi * co;
      b1[M * NC + c] = o;
    }
  }

  // ---------------- stage B : b1 -> b0 ----------------
#pragma unroll
  for (int t = 0; t < R; ++t) {
    const int c2 = 16 * t + lcol;           // column k1*R + n3
    const int k1 = c2 / R;
    const int n3 = c2 % R;
    v2f br[4], bi[4];
#pragma unroll
    for (int j = 0; j < 4; ++j) {
      int K0 = 4 * j + 2 * half;            // row n2
      const v2f f0 = b1[k1 * NC + K0 * R + n3];
      const v2f f1 = b1[k1 * NC + (K0 + 1) * R + n3];
      v2f tr; tr.x = f0.x; tr.y = f1.x; br[j] = tr;
      v2f ti; ti.x = f0.y; ti.y = f1.y; bi[j] = ti;
    }
    v8f crr = {}; v8f cii = {}; v8f cim = {};
#pragma unroll
    for (int j = 0; j < 4; ++j) {
      crr = wmma4(ar[j], br[j], crr);
      cii = wmma4(ai[j], bi[j], cii);
      cim = wmma4(ar[j], bi[j], cim);
      cim = wmma4(ai[j], br[j], cim);
    }
#pragma unroll
    for (int g = 0; g < 8; ++g) {
      const int k2 = g + 8 * half;
      const int e  = (n3 * (k1 + 16 * k2)) % N;
      float s, co; __sincosf(sgn * (TWOPI_F / (float)N) * (float)e, &s, &co);
      const float yr = crr[g] - cii[g];
      const float yi = cim[g];
      v2f o; o.x = yr * co - yi * s; o.y = yr * s + yi * co;
      b0[k2 * NC + c2] = o;
    }
  }

  // ---------------- stage C : radix-R, b0 -> b1 -------
#pragma unroll
  for (int q = 0; q < 8; ++q) {
    const int p  = lane + 32 * q;           // (k1,k2) pair index, 0..255
    const int k1 = p & 15;
    const int k2 = p >> 4;
    const int base = k2 * NC + k1 * R;
    v2f z[R];
#pragma unroll
    for (int n3 = 0; n3 < R; ++n3) z[n3] = b0[base + n3];
    const int k = k1 + 16 * k2;
    if (R == 1) {
      b1[k] = z[0];
    } else if (R == 2) {
      v2f o0; o0.x = z[0].x + z[1].x; o0.y = z[0].y + z[1].y;
      v2f o1; o1.x = z[0].x - z[1].x; o1.y = z[0].y - z[1].y;
      b1[k] = o0; b1[k + 256] = o1;
    } else {
      const float a0r = z[0].x + z[2].x, a0i = z[0].y + z[2].y;
      const float s0r = z[0].x - z[2].x, s0i = z[0].y - z[2].y;
      const float a1r = z[1].x + z[3].x, a1i = z[1].y + z[3].y;
      const float s1r = z[1].x - z[3].x, s1i = z[1].y - z[3].y;
      const float wr = -sgn * s1i;          // (i*SIGN) * (s1r + i s1i)
      const float wi =  sgn * s1r;
      v2f o;
      o.x = a0r + a1r; o.y = a0i + a1i; b1[k]       = o;
      o.x = s0r + wr;  o.y = s0i + wi;  b1[k + 256] = o;
      o.x = a0r - a1r; o.y = a0i - a1i; b1[k + 512] = o;
      o.x = s0r - wr;  o.y = s0i - wi;  b1[k + 768] = o;
    }
  }
}

// ------------------------------- kernels -----------------------------------
// 64 threads = 2 waves per workgroup; each wave owns one row/column plus a
// private LDS slice, so no workgroup barriers are needed.

__global__ __launch_bounds__(64) void kfft_rows(const float* __restrict__ img,
                                                v2f* __restrict__ Fr) {
  __shared__ v2f sm[2][2][1024];
  const int wave = (int)(threadIdx.x >> 5);
  const int lane = (int)(threadIdx.x & 31u);
  lds_v2f* b0 = to_lds(sm[wave][0]);
  lds_v2f* b1 = to_lds(sm[wave][1]);
  const int row = (int)blockIdx.x * 2 + wave;
  const float* src = img + (size_t)row * 1024;
#pragma unroll
  for (int c = 0; c < 8; ++c) {             // coalesced b128 loads, widen r->c
    int idx = c * 128 + lane * 4;
    const float4 v = *reinterpret_cast<const float4*>(src + idx);
    v2f t;
    t.x = v.x; t.y = 0.0f; b0[idx]     = t;
    t.x = v.y;             b0[idx + 1] = t;
    t.x = v.z;             b0[idx + 2] = t;
    t.x = v.w;             b0[idx + 3] = t;
  }
  wave_fft<4, -1>(b0, b1);
  // TDM: contiguous row store (1024 x 8B) straight out of LDS.
  wait_ds0();
  tdm_store(Fr + (size_t)row * 1024, b1, 1024, 1, 1024);
}

__global__ __launch_bounds__(64) void kfft_cols(const v2f* __restrict__ Fin,
                                                v2f* __restrict__ Fout) {
  __shared__ v2f sm[2][2][1024];
  const int wave = (int)(threadIdx.x >> 5);
  lds_v2f* b0 = to_lds(sm[wave][0]);
  lds_v2f* b1 = to_lds(sm[wave][1]);
  const int col = (int)blockIdx.x * 2 + wave;

  // TDM: DMA the strided column (1 x 1024, pitch 1024) into LDS.
  tdm_load(Fin + col, b0, 1, 1024, 1024);
  wait_tensor0();

  wave_fft<4, -1>(b0, b1);

  // TDM: DMA the transformed column back out of LDS.
  wait_ds0();
  tdm_store(Fout + col, b1, 1, 1024, 1024);
  // implicit wait-idle at S_ENDPGM covers the outstanding tensor store
}

// window * spectrum, folded (alias-summed) onto the Nf x Nf grid.
__global__ __launch_bounds__(256) void kwinfold(const v2f* __restrict__ F,
                                                v2f* __restrict__ G,
                                                int scale, int m, int Nf, int d) {
  const int idx = (int)blockIdx.x * 256 + (int)threadIdx.x;
  const int oy = idx / Nf;
  const int ox = idx - oy * Nf;
  float accr = 0.0f, acci = 0.0f;
  for (int a = 0; a < d; ++a) {
    for (int b = 0; b < d; ++b) {
      const int iy = oy + Nf * a;
      const int ix = ox + Nf * b;
      const float w = winval(scale, m, iy, ix);
      if (w != 0.0f) {
        const v2f f = F[(size_t)iy * 1024 + ix];
        accr += w * f.x;
        acci += w * f.y;
      }
    }
  }
  v2f o; o.x = accr; o.y = acci;
  G[idx] = o;
}

template <int R>
__global__ __launch_bounds__(64) void kifft_rows(const v2f* __restrict__ G,
                                                 v2f* __restrict__ T) {
  constexpr int N = 256 * R;
  __shared__ v2f sm[2][2][N];
  const int wave = (int)(threadIdx.x >> 5);
  lds_v2f* b0 = to_lds(sm[wave][0]);
  lds_v2f* b1 = to_lds(sm[wave][1]);
  const int row = (int)blockIdx.x * 2 + wave;

  // TDM: contiguous row load (N x 8B); waited on before the twiddle-matrix setup.
  tdm_load(G + (size_t)row * N, b0, N, 1, N);
  wait_tensor0();

  wave_fft<R, 1>(b0, b1);

  wait_ds0();
  tdm_store(T + (size_t)row * N, b1, N, 1, N);
}

template <int R>
__global__ __launch_bounds__(64) void kifft_cols_out(const v2f* __restrict__ T,
                                                     float* __restrict__ out,
                                                     float sc, size_t off) {
  constexpr int N = 256 * R;
  __shared__ v2f sm[2][2][N];
  const int wave = (int)(threadIdx.x >> 5);
  const int lane = (int)(threadIdx.x & 31u);
  lds_v2f* b0 = to_lds(sm[wave][0]);
  lds_v2f* b1 = to_lds(sm[wave][1]);
  const int col = (int)blockIdx.x * 2 + wave;

  // TDM: DMA the strided column (1 x N, pitch N) into LDS.
  tdm_load(T + col, b0, 1, N, N);
  wait_tensor0();

  wave_fft<R, 1>(b0, b1);

#pragma unroll
  for (int c = 0; c < N / 32; ++c) {
    int k = c * 32 + lane;
    const v2f v = b1[k];
    out[off + (size_t)k * N + col] = v.x * sc;
  }
}

// ------------------------------- launcher ----------------------------------

extern "C" void kernel_launch(void* const* d_in, const int* in_sizes, int n_in,
                              void* d_out, int out_size, void* d_ws, size_t ws_size,
                              hipStream_t stream) {
  (void)in_sizes; (void)n_in; (void)out_size; (void)ws_size;
  const float* img = (const float*)d_in[0];
  float* out = (float*)d_out;

  v2f* F = (v2f*)d_ws;                  // full spectrum      (8 MB)
  v2f* G = F + 1024 * 1024;             // folded band spec   (8 MB)
  v2f* T = G + 1024 * 1024;             // row-FFT temp       (8 MB)

  // forward 2-D FFT: rows (image -> T), then columns (T -> F)
  kfft_rows<<<512, 64, 0, stream>>>(img, T);
  kfft_cols<<<512, 64, 0, stream>>>(T, F);

  size_t off = 0;

  // scale 0: lowpass, N=256, d=4
  {
    const int N = 256, d = 4;
    kwinfold<<<(N * N) / 256, 256, 0, stream>>>(F, G, 0, 0, N, d);
    kifft_rows<1><<<N / 2, 64, 0, stream>>>(G, T);
    kifft_cols_out<1><<<N / 2, 64, 0, stream>>>(T, out,
        1.0f / ((float)N * (float)N * (float)d), off);
    off += (size_t)N * N;
  }
  // scale 1: 6 wedges, N=512, d=2
  for (int m = 0; m < 6; ++m) {
    const int N = 512, d = 2;
    kwinfold<<<(N * N) / 256, 256, 0, stream>>>(F, G, 1, m, N, d);
    kifft_rows<2><<<N / 2, 64, 0, stream>>>(G, T);
    kifft_cols_out<2><<<N / 2, 64, 0, stream>>>(T, out,
        1.0f / ((float)N * (float)N * (float)d), off);
    off += (size_t)N * N;
  }
  // scale 2: 12 wedges, N=1024, d=1
  for (int m = 0; m < 12; ++m) {
    const int N = 1024, d = 1;
    kwinfold<<<(N * N) / 256, 256, 0, stream>>>(F, G, 2, m, N, d);
    kifft_rows<4><<<N / 2, 64, 0, stream>>>(G, T);
    kifft_cols_out<4><<<N / 2, 64, 0, stream>>>(T, out,
        1.0f / ((float)N * (float)N * (float)d), off);
    off += (size_t)N * N;
  }
}